// Model2d3d_60619168415827
// MI455X (gfx1250) — compile-verified
//
#include <hip/hip_runtime.h>
#include <math.h>

#define BB 16
#define VV 5
#define NN 8192
#define CC 128
#define HWP (41 * 32)   // 1312
#define NP1 (NN + 1)    // 8193
#define PPB 4           // points per block in phase 2
#define STG (PPB * VV)  // 20 staged (point, view) pairs per block

// ---------------------------------------------------------------------------
// Kernel 0: init winner[] = -1
// ---------------------------------------------------------------------------
__global__ void k_init_winner(int* __restrict__ winner, int n) {
    int t = blockIdx.x * blockDim.x + threadIdx.x;
    if (t < n) winner[t] = -1;
}

// ---------------------------------------------------------------------------
// Kernel 1: last-writer-wins scatter resolution.
// winner[bv][tgt] = max i such that i < count and ind3d[bv][1+i] == tgt.
// ---------------------------------------------------------------------------
__global__ void k_phase1(const int* __restrict__ ind3d, int* __restrict__ winner) {
    int t = blockIdx.x * blockDim.x + threadIdx.x;
    if (t >= BB * VV * NN) return;
    int bv = t / NN;
    int i  = t - bv * NN;
    int count = ind3d[(size_t)bv * NP1];          // element 0 = valid count
    if (i < count) {
        int tgt = ind3d[(size_t)bv * NP1 + 1 + i];
        if ((unsigned)tgt < (unsigned)NN)
            atomicMax(&winner[bv * NN + tgt], i);
    }
}

// ---------------------------------------------------------------------------
// Kernel 2: one 128-thread block per 4 points of one batch. Stage the 20
// (point, view) winners and then the 20 source indices into LDS with CDNA5
// async global->LDS loads (ASYNCcnt path), then gather 128 channels per
// point from the L2-resident feature maps, max over views (implicit zero for
// views that did not scatter into the slot), and write fused output rows.
// ---------------------------------------------------------------------------
__global__ __launch_bounds__(128) void k_phase2(
    const float* __restrict__ pc,     // (B, N, 3)
    const float* __restrict__ feat,   // (BV, C, H*W)
    const int*   __restrict__ ind2d,  // (BV, N+1)
    const int*   __restrict__ winner, // (BV, N)
    float*       __restrict__ out)    // (B, N, 3+C)
{
    __shared__ int s_w[STG];
    __shared__ int s_src[STG];

    const int n0  = blockIdx.x * PPB;
    const int b   = blockIdx.y;
    const int tid = threadIdx.x;

    // --- round 1: stage winners, global -> LDS via async data mover ---
    if (tid < STG) {
        const int p  = tid / VV;
        const int v  = tid - p * VV;
        const int bv = b * VV + v;
        unsigned lds_off = (unsigned)(size_t)&s_w[tid];
        const int* gaddr = &winner[bv * NN + (n0 + p)];
        asm volatile("global_load_async_to_lds_b32 %0, %1, off"
                     :: "v"(lds_off), "v"(gaddr) : "memory");
    }
    asm volatile("s_wait_asynccnt 0" ::: "memory");
    __syncthreads();

    // --- round 2: stage 2d source indices for (point, view) with a winner ---
    if (tid < STG) {
        const int w = s_w[tid];
        if (w >= 0) {
            const int v  = tid % VV;
            const int bv = b * VV + v;
            unsigned lds_off = (unsigned)(size_t)&s_src[tid];
            const int* gaddr = &ind2d[(size_t)bv * NP1 + 1 + w];
            asm volatile("global_load_async_to_lds_b32 %0, %1, off"
                         :: "v"(lds_off), "v"(gaddr) : "memory");
        }
    }
    asm volatile("s_wait_asynccnt 0" ::: "memory");
    __syncthreads();

    // --- per-channel max over views for each of the PPB points ---
    const int c = tid;                 // 128 threads == 128 channels
#pragma unroll
    for (int p = 0; p < PPB; ++p) {
        float vmax = -INFINITY;
#pragma unroll
        for (int v = 0; v < VV; ++v) {
            float val = 0.0f;          // un-scattered slot contributes 0
            if (s_w[p * VV + v] >= 0) {
                const int bv = b * VV + v;
                const int s  = s_src[p * VV + v];
                val = feat[((size_t)(bv * CC + c)) * HWP + (size_t)s];
            }
            vmax = fmaxf(vmax, val);
        }
        const size_t row = (size_t)(b * NN + n0 + p);
        float* orow = out + row * (size_t)(3 + CC);
        orow[3 + c] = vmax;
        if (tid < 3) orow[tid] = pc[row * 3 + tid];
    }
}

// ---------------------------------------------------------------------------
// Launcher
// ---------------------------------------------------------------------------
extern "C" void kernel_launch(void* const* d_in, const int* in_sizes, int n_in,
                              void* d_out, int out_size, void* d_ws, size_t ws_size,
                              hipStream_t stream) {
    const float* pc    = (const float*)d_in[0];  // (16, 8192, 3)
    const float* feat  = (const float*)d_in[1];  // (80, 128, 41, 32)
    const int*   ind3d = (const int*)d_in[2];    // (80, 8193)
    const int*   ind2d = (const int*)d_in[3];    // (80, 8193)
    float*       out   = (float*)d_out;          // (16, 8192, 131)

    int* winner = (int*)d_ws;                    // 80*8192 ints = 2.62 MB
    const int nw = BB * VV * NN;

    k_init_winner<<<(nw + 255) / 256, 256, 0, stream>>>(winner, nw);
    k_phase1<<<(nw + 255) / 256, 256, 0, stream>>>(ind3d, winner);

    dim3 grid(NN / PPB, BB);
    k_phase2<<<grid, 128, 0, stream>>>(pc, feat, ind2d, winner, out);
}